// ChamferDistanceLoss_45552423141595
// MI455X (gfx1250) — compile-verified
//
#include <hip/hip_runtime.h>
#include <hip/hip_bf16.h>
#include <math.h>

// Chamfer distance on MI455X (gfx1250, wave32).
// d2(a,b) = |a|^2 - 2 a.b + |b|^2 as a K=4 f32 WMMA:
//   A row (16x4)  = [ax, ay, az, 1]
//   B col (4x16)  = [-2bx, -2by, -2bz, |b|^2]   (pre-transformed once)
//   D tile        = -2 a.b + |b|^2   (|a|^2 added after the min over columns)

typedef float v2f __attribute__((ext_vector_type(2)));
typedef float v8f __attribute__((ext_vector_type(8)));

#define BATCH        4
#define THREADS      256        // 8 waves
#define TILES_PER_WAVE 2        // 32 rows per wave
#define ROWS_PER_WG  (THREADS / 32 * 16 * TILES_PER_WAVE)   // 256
#define CHUNK        2048       // reference points staged in LDS (32 KB)

// Raw min ops: inputs are provably finite here, so skip clang's IEEE
// sNaN canonicalization (v_max_num_f32 x,x self-ops) around fminf.
__device__ __forceinline__ float fmin_asm(float a, float b) {
    float r;
    asm("v_min_num_f32 %0, %1, %2" : "=v"(r) : "v"(a), "v"(b));
    return r;
}
__device__ __forceinline__ float fmin3_asm(float a, float b, float c) {
    float r;
    asm("v_min3_num_f32 %0, %1, %2, %3" : "=v"(r) : "v"(a), "v"(b), "v"(c));
    return r;
}

// ds_swizzle group-of-32: and=0x1f, or=0, xor=IMM>>10
template <int IMM>
__device__ __forceinline__ float swz_min(float v) {
    int s = __builtin_amdgcn_ds_swizzle(__float_as_int(v), IMM);
    return fmin_asm(v, __int_as_float(s));
}
__device__ __forceinline__ float group16_min(float v) {
    v = swz_min<0x041f>(v);   // SWAPX1
    v = swz_min<0x081f>(v);   // SWAPX2
    v = swz_min<0x101f>(v);   // SWAPX4
    v = swz_min<0x201f>(v);   // SWAPX8
    return v;
}

// Transform raw points [total,3] -> B-operand layout [total,4]:
//   (-2x, -2y, -2z, x^2+y^2+z^2)
__global__ __launch_bounds__(THREADS)
void chamfer_prep_kernel(const float* __restrict__ p, float4* __restrict__ t, int total)
{
    int i = blockIdx.x * THREADS + threadIdx.x;
    if (i < total) {
        const float x = p[(size_t)i * 3 + 0];
        const float y = p[(size_t)i * 3 + 1];
        const float z = p[(size_t)i * 3 + 2];
        t[i] = make_float4(-2.0f * x, -2.0f * y, -2.0f * z, x * x + y * y + z * z);
    }
}

__global__ __launch_bounds__(THREADS)
void chamfer_nn_min_kernel(const float* __restrict__ A,    // [B,N,3] raw query points
                           const float4* __restrict__ Bt,  // [B,M,4] transformed refs
                           float* __restrict__ outMin,     // [B,N] min squared distance
                           int N, int M)
{
    __shared__ float lds[CHUNK * 4];

    const int batch   = blockIdx.y;
    const int tid     = threadIdx.x;
    const int wave    = tid >> 5;
    const int lane    = tid & 31;
    const int half    = lane >> 4;      // 0: lanes 0-15, 1: lanes 16-31
    const int l16     = lane & 15;
    const int rowbase = blockIdx.x * ROWS_PER_WG + wave * (16 * TILES_PER_WAVE);

    // ---- A operands: rows rowbase+l16 and rowbase+16+l16 (16x4 f32 A layout:
    //  lanes 0-15 hold K0,K1 = x,y ; lanes 16-31 hold K2,K3 = z,1)
    v2f   avec[TILES_PER_WAVE];
    float a2[TILES_PER_WAVE];
#pragma unroll
    for (int t = 0; t < TILES_PER_WAVE; ++t) {
        const int row = rowbase + t * 16 + l16;
        const float* ap = A + ((size_t)batch * N + row) * 3;
        const float ax = ap[0], ay = ap[1], az = ap[2];
        a2[t] = ax * ax + ay * ay + az * az;
        avec[t].x = half ? az   : ax;
        avec[t].y = half ? 1.0f : ay;
    }

    v8f acc[TILES_PER_WAVE];
#pragma unroll
    for (int t = 0; t < TILES_PER_WAVE; ++t)
#pragma unroll
        for (int k = 0; k < 8; ++k) acc[t][k] = 3.4e38f;

    const float4* bbase = Bt + (size_t)batch * M;

    for (int chunk = 0; chunk < M; chunk += CHUNK) {
        __syncthreads();
        {   // cooperative stage: CHUNK float4s via b128
            float4* s4 = (float4*)lds;
            const float4* g4 = bbase + chunk;
#pragma unroll
            for (int t = tid; t < CHUNK; t += THREADS)
                s4[t] = g4[t];
        }
        __syncthreads();

        // Two column-tiles per step: one ds_load_2addr_b64, 4 WMMAs,
        // 16 v_min3 (acc, d0, d1) with no canonicalization.
#pragma unroll 2
        for (int jt = 0; jt < CHUNK; jt += 32) {
            // B operand 4x16: lanes 0-15 take words 0-1 (-2bx,-2by),
            // lanes 16-31 take words 2-3 (-2bz,|b|^2).
            const v2f b0 = *(const v2f*)&lds[(jt +      l16) * 4 + half * 2];
            const v2f b1 = *(const v2f*)&lds[(jt + 16 + l16) * 4 + half * 2];
#pragma unroll
            for (int t = 0; t < TILES_PER_WAVE; ++t) {
                v8f cz0 = {}, cz1 = {};
                v8f d0 = __builtin_amdgcn_wmma_f32_16x16x4_f32(
                             false, avec[t], false, b0, (short)0, cz0, false, false);
                v8f d1 = __builtin_amdgcn_wmma_f32_16x16x4_f32(
                             false, avec[t], false, b1, (short)0, cz1, false, false);
#pragma unroll
                for (int k = 0; k < 8; ++k)
                    acc[t][k] = fmin3_asm(acc[t][k], d0[k], d1[k]);
            }
        }
    }

    // Min across the 16 lanes of each half via ds_swizzle SWAPX1/2/4/8.
#pragma unroll
    for (int t = 0; t < TILES_PER_WAVE; ++t)
#pragma unroll
        for (int k = 0; k < 8; ++k)
            acc[t][k] = group16_min(acc[t][k]);

    // D slot k on half h holds row (k + 8h). Lane with (l16>>3)==half owns
    // rows rowbase + t*16 + l16 and already has their |a|^2 in a2[t].
    if ((l16 >> 3) == half) {
        const int k = l16 & 7;
#pragma unroll
        for (int t = 0; t < TILES_PER_WAVE; ++t) {
            float m = acc[t][0];
#pragma unroll
            for (int kk = 1; kk < 8; ++kk)
                if (k == kk) m = acc[t][kk];
            const float d2 = fmaxf(m + a2[t], 0.0f);
            outMin[(size_t)batch * N + rowbase + t * 16 + l16] = d2;
        }
    }
}

// Sum sqrt(min d2) over both directions and all batches; divide by count.
// Valid because N == M: mean_b((mean+mean)/2) == sum_all / (2*B*N).
__global__ __launch_bounds__(THREADS)
void chamfer_reduce_kernel(const float* __restrict__ mins, float* __restrict__ out,
                           int total)
{
    __shared__ float red[THREADS];
    float s = 0.0f;
    for (int i = threadIdx.x; i < total; i += THREADS)
        s += sqrtf(mins[i]);
    red[threadIdx.x] = s;
    __syncthreads();
    for (int off = THREADS / 2; off > 0; off >>= 1) {
        if (threadIdx.x < off) red[threadIdx.x] += red[threadIdx.x + off];
        __syncthreads();
    }
    if (threadIdx.x == 0) out[0] = red[0] / (float)total;
}

extern "C" void kernel_launch(void* const* d_in, const int* in_sizes, int n_in,
                              void* d_out, int out_size, void* d_ws, size_t ws_size,
                              hipStream_t stream)
{
    const float* tmpl = (const float*)d_in[0];   // template [B,N,3] fp32
    const float* src  = (const float*)d_in[1];   // source   [B,M,3] fp32
    float* out = (float*)d_out;

    const int Bn = BATCH;
    const int N  = in_sizes[0] / (Bn * 3);       // 8192
    const int M  = in_sizes[1] / (Bn * 3);       // 8192

    // Workspace layout (floats): tA[B*N*4] | tB[B*M*4] | w0[B*N] | w1[B*M]
    float4* tA = (float4*)d_ws;
    float4* tB = tA + (size_t)Bn * N;
    float*  w0 = (float*)(tB + (size_t)Bn * M);
    float*  w1 = w0 + (size_t)Bn * N;

    dim3 block(THREADS);

    const int totA = Bn * N, totB = Bn * M;
    chamfer_prep_kernel<<<dim3((totA + THREADS - 1) / THREADS), block, 0, stream>>>(
        tmpl, tA, totA);
    chamfer_prep_kernel<<<dim3((totB + THREADS - 1) / THREADS), block, 0, stream>>>(
        src, tB, totB);

    chamfer_nn_min_kernel<<<dim3(N / ROWS_PER_WG, Bn), block, 0, stream>>>(
        tmpl, tB, w0, N, M);
    chamfer_nn_min_kernel<<<dim3(M / ROWS_PER_WG, Bn), block, 0, stream>>>(
        src, tA, w1, M, N);

    chamfer_reduce_kernel<<<1, THREADS, 0, stream>>>(w0, out, Bn * (N + M));
}